// GuidanceConstruct_8254927142934
// MI455X (gfx1250) — compile-verified
//
#include <hip/hip_runtime.h>
#include <math.h>

// Problem constants (from reference)
#define EPSf 1e-5f
#define SCALEf 0.35355339059327373f   // 1/sqrt(8)
#define NTOT 4194304                  // B*C*K*L = 2*64*256*128

typedef __attribute__((ext_vector_type(2))) float v2f;
typedef __attribute__((ext_vector_type(8))) float v8f;

// ---------------- WMMA f32 16x16x4 core ----------------
static __device__ __forceinline__ v8f wmma4(v2f a, v2f b, v8f c) {
  return __builtin_amdgcn_wmma_f32_16x16x4_f32(false, a, false, b, (short)0, c, false, false);
}

// A fragment: A row-major (M x K), element [tm+m, k0+ko]
static __device__ __forceinline__ v2f frag_a(const float* A, int lda, int tm, int k0, int lane) {
  const int m = lane & 15;
  const int ko = (lane >> 4) << 1;
  const float* p = A + (tm + m) * lda + k0 + ko;
  v2f r; r.x = p[0]; r.y = p[1];
  return r;
}
// A from transposed storage (stored K x M): element [tm+m, k] = At[k*ldat + tm+m]
static __device__ __forceinline__ v2f frag_aT(const float* At, int ldat, int tm, int k0, int lane) {
  const int m = lane & 15;
  const int ko = (lane >> 4) << 1;
  v2f r;
  r.x = At[(k0 + ko) * ldat + tm + m];
  r.y = At[(k0 + ko + 1) * ldat + tm + m];
  return r;
}
// B fragment: B row-major (K x N), element [k0+kr, tn+n]
static __device__ __forceinline__ v2f frag_b(const float* Bm, int ldb, int tn, int k0, int lane) {
  const int n = lane & 15;
  const int kr = (lane >> 4) << 1;
  v2f r;
  r.x = Bm[(k0 + kr) * ldb + tn + n];
  r.y = Bm[(k0 + kr + 1) * ldb + tn + n];
  return r;
}
// B from transposed storage (stored N x K): element [k, tn+n] = Bt[(tn+n)*ldbt + k]
static __device__ __forceinline__ v2f frag_bT(const float* Bt, int ldbt, int tn, int k0, int lane) {
  const int n = lane & 15;
  const int kr = (lane >> 4) << 1;
  const float* p = Bt + (tn + n) * ldbt + k0 + kr;
  v2f r; r.x = p[0]; r.y = p[1];
  return r;
}

// 8-wave (256-thread) block GEMM over 16x16 tiles; FA/FB produce fragments, FS stores D tiles.
template <class FA, class FB, class FS>
static __device__ __forceinline__ void gemm_block(int M, int N, int K, FA fa, FB fb, FS fs) {
  const int lane = threadIdx.x & 31;
  const int wave = threadIdx.x >> 5;
  const int mt = M >> 4;
  const int nt = N >> 4;
  const int tiles = mt * nt;
  for (int t = wave; t < tiles; t += 8) {
    const int tm = (t % mt) << 4;
    const int tn = (t / mt) << 4;
    v8f acc = {};
    for (int k0 = 0; k0 < K; k0 += 4)
      acc = wmma4(fa(tm, k0, lane), fb(tn, k0, lane), acc);
    fs(tm, tn, acc, lane);
  }
}

// ---------------- adaptive adjacency: adpT = softmax(relu(n1@n2), axis=1)^T ----------------
__global__ __launch_bounds__(256) void k_adp(const float* n1, const float* n2, float* adpT) {
  const int w = blockIdx.x;     // row (softmax over columns)
  const int v = threadIdx.x;    // col
  float s = 0.f;
#pragma unroll
  for (int d = 0; d < 10; ++d) s += n1[w * 10 + d] * n2[d * 256 + v];
  s = fmaxf(s, 0.f);
  __shared__ float red[256];
  red[v] = s; __syncthreads();
  for (int off = 128; off > 0; off >>= 1) {
    if (v < off) red[v] = fmaxf(red[v], red[v + off]);
    __syncthreads();
  }
  const float m = red[0]; __syncthreads();
  const float e = expf(s - m);
  red[v] = e; __syncthreads();
  for (int off = 128; off > 0; off >>= 1) {
    if (v < off) red[v] += red[v + off];
    __syncthreads();
  }
  adpT[v * 256 + w] = e / red[0];
}

__global__ __launch_bounds__(256) void k_tr256(const float* src, float* dst) {
  dst[threadIdx.x * 256 + blockIdx.x] = src[blockIdx.x * 256 + threadIdx.x];
}

// ---------------- AdaptiveGCN: one block per (b,l) slice ----------------
#define GLD 257
__global__ __launch_bounds__(256) void k_gcn(const float* y, const float* a1T, const float* a2T,
                                             const float* adpT, const float* gw, const float* gb,
                                             float* yl) {
  __shared__ float X[64 * GLD];
  __shared__ float B1s[64 * GLD];
  __shared__ float B2s[64 * GLD];
  const int b = blockIdx.x >> 7;
  const int l = blockIdx.x & 127;
  const int lane = threadIdx.x & 31;
  const int wave = threadIdx.x >> 5;
  for (int i = threadIdx.x; i < 64 * 256; i += 256) {
    const int c = i >> 8, v = i & 255;
    X[c * GLD + v] = y[((((b << 6) + c) << 8) | v) * 128 + l];
  }
  __syncthreads();
  v8f acc[8];
  {
    v8f z = {};
#pragma unroll
    for (int t = 0; t < 8; ++t) acc[t] = z;
  }
  // accumulate  g += gcn_w[:, j*64:(j+1)*64] @ S   (S: 64x256 in LDS)
  auto accum = [&](const float* S, int j) {
#pragma unroll
    for (int t = 0; t < 8; ++t) {
      const int tm = (t & 3) << 4;
      const int tn = (((wave << 1) | (t >> 2))) << 4;
      v8f a = acc[t];
      for (int k0 = 0; k0 < 64; k0 += 4)
        a = wmma4(frag_a(gw + j * 64, 448, tm, k0, lane), frag_b(S, GLD, tn, k0, lane), a);
      acc[t] = a;
    }
  };
  accum(X, 0);
  const float* As[3] = {a1T, a2T, adpT};
  for (int ai = 0; ai < 3; ++ai) {
    __syncthreads();
    // B1 = X @ A^T   (A^T stored row-major as aT: 256x256)
    gemm_block(64, 256, 256,
      [&](int tm, int k0, int ln) { return frag_a(X, GLD, tm, k0, ln); },
      [&](int tn, int k0, int ln) { return frag_b(As[ai], 256, tn, k0, ln); },
      [&](int tm, int tn, v8f av, int ln) {
        const int n = ln & 15, mh = (ln >> 4) << 3;
#pragma unroll
        for (int r = 0; r < 8; ++r) B1s[(tm + mh + r) * GLD + tn + n] = av[r];
      });
    __syncthreads();
    accum(B1s, 1 + 2 * ai);
    // B2 = B1 @ A^T
    gemm_block(64, 256, 256,
      [&](int tm, int k0, int ln) { return frag_a(B1s, GLD, tm, k0, ln); },
      [&](int tn, int k0, int ln) { return frag_b(As[ai], 256, tn, k0, ln); },
      [&](int tm, int tn, v8f av, int ln) {
        const int n = ln & 15, mh = (ln >> 4) << 3;
#pragma unroll
        for (int r = 0; r < 8; ++r) B2s[(tm + mh + r) * GLD + tn + n] = av[r];
      });
    __syncthreads();
    accum(B2s, 2 + 2 * ai);
  }
  // write g + bias -> yl[b, o, k, l]
#pragma unroll
  for (int t = 0; t < 8; ++t) {
    const int tm = (t & 3) << 4;
    const int tn = (((wave << 1) | (t >> 2))) << 4;
    const int n = lane & 15, mh = (lane >> 4) << 3;
#pragma unroll
    for (int r = 0; r < 8; ++r) {
      const int o = tm + mh + r, k = tn + n;
      yl[((((b << 6) + o) << 8) | k) * 128 + l] = acc[t][r] + gb[o];
    }
  }
}

// ---------------- Linformer spatial attention: one block per (b,l) ----------------
#define SLD 65
#define PLD 81
__global__ __launch_bounds__(256) void k_spatial(const float* y, const float* wq, const float* wk,
                                                 const float* wv, const float* pk, const float* pv,
                                                 const float* wo, const float* bo, float* ys) {
  __shared__ float XS[256 * SLD];   // xs, later reused as O
  __shared__ float Qs[256 * SLD];
  __shared__ float Ts[256 * SLD];   // temp for k_, v_, then per-head S
  __shared__ float KP[64 * PLD];
  __shared__ float VP[64 * PLD];
  const int b = blockIdx.x >> 7;
  const int l = blockIdx.x & 127;
  for (int i = threadIdx.x; i < 64 * 256; i += 256) {
    const int c = i >> 8, n = i & 255;
    XS[n * SLD + c] = y[((((b << 6) + c) << 8) | n) * 128 + l];
  }
  for (int i = threadIdx.x; i < 64 * (PLD - 64); i += 256) {
    const int p = i / (PLD - 64), c = 64 + i % (PLD - 64);
    VP[p * PLD + c] = 0.f;   // zero pad for 16-wide head tiles
  }
  __syncthreads();
  // Q = xs @ wq ; Ts = xs @ wk
  gemm_block(256, 64, 64,
    [&](int tm, int k0, int ln) { return frag_a(XS, SLD, tm, k0, ln); },
    [&](int tn, int k0, int ln) { return frag_b(wq, 64, tn, k0, ln); },
    [&](int tm, int tn, v8f av, int ln) {
      const int n = ln & 15, mh = (ln >> 4) << 3;
#pragma unroll
      for (int r = 0; r < 8; ++r) Qs[(tm + mh + r) * SLD + tn + n] = av[r];
    });
  gemm_block(256, 64, 64,
    [&](int tm, int k0, int ln) { return frag_a(XS, SLD, tm, k0, ln); },
    [&](int tn, int k0, int ln) { return frag_b(wk, 64, tn, k0, ln); },
    [&](int tm, int tn, v8f av, int ln) {
      const int n = ln & 15, mh = (ln >> 4) << 3;
#pragma unroll
      for (int r = 0; r < 8; ++r) Ts[(tm + mh + r) * SLD + tn + n] = av[r];
    });
  __syncthreads();
  // KP = pk^T @ k_
  gemm_block(64, 64, 256,
    [&](int tm, int k0, int ln) { return frag_aT(pk, 64, tm, k0, ln); },
    [&](int tn, int k0, int ln) { return frag_b(Ts, SLD, tn, k0, ln); },
    [&](int tm, int tn, v8f av, int ln) {
      const int n = ln & 15, mh = (ln >> 4) << 3;
#pragma unroll
      for (int r = 0; r < 8; ++r) KP[(tm + mh + r) * PLD + tn + n] = av[r];
    });
  __syncthreads();
  // Ts = xs @ wv
  gemm_block(256, 64, 64,
    [&](int tm, int k0, int ln) { return frag_a(XS, SLD, tm, k0, ln); },
    [&](int tn, int k0, int ln) { return frag_b(wv, 64, tn, k0, ln); },
    [&](int tm, int tn, v8f av, int ln) {
      const int n = ln & 15, mh = (ln >> 4) << 3;
#pragma unroll
      for (int r = 0; r < 8; ++r) Ts[(tm + mh + r) * SLD + tn + n] = av[r];
    });
  __syncthreads();
  // VP = pv^T @ v_
  gemm_block(64, 64, 256,
    [&](int tm, int k0, int ln) { return frag_aT(pv, 64, tm, k0, ln); },
    [&](int tn, int k0, int ln) { return frag_b(Ts, SLD, tn, k0, ln); },
    [&](int tm, int tn, v8f av, int ln) {
      const int n = ln & 15, mh = (ln >> 4) << 3;
#pragma unroll
      for (int r = 0; r < 8; ++r) VP[(tm + mh + r) * PLD + tn + n] = av[r];
    });
  __syncthreads();
  for (int h = 0; h < 8; ++h) {
    // S = Q_h @ KP_h^T   (256x64, K=8)
    gemm_block(256, 64, 8,
      [&](int tm, int k0, int ln) { return frag_a(Qs + h * 8, SLD, tm, k0, ln); },
      [&](int tn, int k0, int ln) { return frag_bT(KP + h * 8, PLD, tn, k0, ln); },
      [&](int tm, int tn, v8f av, int ln) {
        const int n = ln & 15, mh = (ln >> 4) << 3;
#pragma unroll
        for (int r = 0; r < 8; ++r) Ts[(tm + mh + r) * SLD + tn + n] = av[r];
      });
    __syncthreads();
    {
      const int t = threadIdx.x;  // one row per thread
      float mx = -1e30f;
      for (int j = 0; j < 64; ++j) mx = fmaxf(mx, Ts[t * SLD + j] * SCALEf);
      float sum = 0.f;
      for (int j = 0; j < 64; ++j) {
        const float e = expf(Ts[t * SLD + j] * SCALEf - mx);
        Ts[t * SLD + j] = e; sum += e;
      }
      const float inv = 1.f / sum;
      for (int j = 0; j < 64; ++j) Ts[t * SLD + j] *= inv;
    }
    __syncthreads();
    // O_h = S @ VP[:, h*8 : h*8+16)  (keep first 8 cols) -> XS
    gemm_block(256, 16, 64,
      [&](int tm, int k0, int ln) { return frag_a(Ts, SLD, tm, k0, ln); },
      [&](int tn, int k0, int ln) { return frag_b(VP + h * 8, PLD, tn, k0, ln); },
      [&](int tm, int tn, v8f av, int ln) {
        const int n = ln & 15, mh = (ln >> 4) << 3;
        if (n < 8) {
#pragma unroll
          for (int r = 0; r < 8; ++r) XS[(tm + mh + r) * SLD + h * 8 + n] = av[r];
        }
      });
    __syncthreads();
  }
  // out = O @ wo + bo  -> ys[b, c, n, l]
  gemm_block(256, 64, 64,
    [&](int tm, int k0, int ln) { return frag_a(XS, SLD, tm, k0, ln); },
    [&](int tn, int k0, int ln) { return frag_b(wo, 64, tn, k0, ln); },
    [&](int tm, int tn, v8f av, int ln) {
      const int n = ln & 15, mh = (ln >> 4) << 3;
#pragma unroll
      for (int r = 0; r < 8; ++r) {
        const int tok = tm + mh + r, c = tn + n;
        ys[((((b << 6) + c) << 8) | tok) * 128 + l] = av[r] + bo[c];
      }
    });
}

// ---------------- temporal transformer layer: one block per (b,k) ----------------
#define TLD 65
#define VLD 81
#define STLD 129
__global__ __launch_bounds__(256) void k_temporal(const float* y,
    const float* wq, const float* wk, const float* wv,
    const float* bq, const float* bk, const float* bv,
    const float* wo, const float* bo,
    const float* l1w, const float* l1b, const float* l2w, const float* l2b,
    const float* ln1g, const float* ln1b, const float* ln2g, const float* ln2b,
    float* yt) {
  __shared__ float XT[128 * TLD];
  __shared__ float QT[128 * TLD];
  __shared__ float KT[128 * TLD];
  __shared__ float VT[128 * VLD];
  __shared__ float ST[128 * STLD];   // per-head S; later reused (ld TLD) as src
  __shared__ float OT[128 * TLD];
  const int b = blockIdx.x >> 8;
  const int k = blockIdx.x & 255;
  for (int i = threadIdx.x; i < 64 * 128; i += 256) {
    const int c = i >> 7, l = i & 127;
    XT[l * TLD + c] = y[((((b << 6) + c) << 8) | k) * 128 + l];
  }
  for (int i = threadIdx.x; i < 128 * (VLD - 64); i += 256) {
    const int l = i / (VLD - 64), c = 64 + i % (VLD - 64);
    VT[l * VLD + c] = 0.f;
  }
  __syncthreads();
  gemm_block(128, 64, 64,
    [&](int tm, int k0, int ln) { return frag_a(XT, TLD, tm, k0, ln); },
    [&](int tn, int k0, int ln) { return frag_b(wq, 64, tn, k0, ln); },
    [&](int tm, int tn, v8f av, int ln) {
      const int n = ln & 15, mh = (ln >> 4) << 3;
#pragma unroll
      for (int r = 0; r < 8; ++r) QT[(tm + mh + r) * TLD + tn + n] = av[r] + bq[tn + n];
    });
  gemm_block(128, 64, 64,
    [&](int tm, int k0, int ln) { return frag_a(XT, TLD, tm, k0, ln); },
    [&](int tn, int k0, int ln) { return frag_b(wk, 64, tn, k0, ln); },
    [&](int tm, int tn, v8f av, int ln) {
      const int n = ln & 15, mh = (ln >> 4) << 3;
#pragma unroll
      for (int r = 0; r < 8; ++r) KT[(tm + mh + r) * TLD + tn + n] = av[r] + bk[tn + n];
    });
  gemm_block(128, 64, 64,
    [&](int tm, int k0, int ln) { return frag_a(XT, TLD, tm, k0, ln); },
    [&](int tn, int k0, int ln) { return frag_b(wv, 64, tn, k0, ln); },
    [&](int tm, int tn, v8f av, int ln) {
      const int n = ln & 15, mh = (ln >> 4) << 3;
#pragma unroll
      for (int r = 0; r < 8; ++r) VT[(tm + mh + r) * VLD + tn + n] = av[r] + bv[tn + n];
    });
  __syncthreads();
  for (int h = 0; h < 8; ++h) {
    gemm_block(128, 128, 8,
      [&](int tm, int k0, int ln) { return frag_a(QT + h * 8, TLD, tm, k0, ln); },
      [&](int tn, int k0, int ln) { return frag_bT(KT + h * 8, TLD, tn, k0, ln); },
      [&](int tm, int tn, v8f av, int ln) {
        const int n = ln & 15, mh = (ln >> 4) << 3;
#pragma unroll
        for (int r = 0; r < 8; ++r) ST[(tm + mh + r) * STLD + tn + n] = av[r];
      });
    __syncthreads();
    if (threadIdx.x < 128) {
      const int t = threadIdx.x;
      float mx = -1e30f;
      for (int j = 0; j < 128; ++j) mx = fmaxf(mx, ST[t * STLD + j] * SCALEf);
      float sum = 0.f;
      for (int j = 0; j < 128; ++j) {
        const float e = expf(ST[t * STLD + j] * SCALEf - mx);
        ST[t * STLD + j] = e; sum += e;
      }
      const float inv = 1.f / sum;
      for (int j = 0; j < 128; ++j) ST[t * STLD + j] *= inv;
    }
    __syncthreads();
    gemm_block(128, 16, 128,
      [&](int tm, int k0, int ln) { return frag_a(ST, STLD, tm, k0, ln); },
      [&](int tn, int k0, int ln) { return frag_b(VT + h * 8, VLD, tn, k0, ln); },
      [&](int tm, int tn, v8f av, int ln) {
        const int n = ln & 15, mh = (ln >> 4) << 3;
        if (n < 8) {
#pragma unroll
          for (int r = 0; r < 8; ++r) OT[(tm + mh + r) * TLD + h * 8 + n] = av[r];
        }
      });
    __syncthreads();
  }
  // src = xt + O@wo + bo  (into ST region, ld TLD)
  float* SRC = ST;
  gemm_block(128, 64, 64,
    [&](int tm, int k0, int ln) { return frag_a(OT, TLD, tm, k0, ln); },
    [&](int tn, int k0, int ln) { return frag_b(wo, 64, tn, k0, ln); },
    [&](int tm, int tn, v8f av, int ln) {
      const int n = ln & 15, mh = (ln >> 4) << 3;
#pragma unroll
      for (int r = 0; r < 8; ++r) {
        const int row = tm + mh + r, c = tn + n;
        SRC[row * TLD + c] = av[r] + bo[c] + XT[row * TLD + c];
      }
    });
  __syncthreads();
  if (threadIdx.x < 128) {   // LN1
    const int t = threadIdx.x;
    float s = 0.f, s2 = 0.f;
    for (int c = 0; c < 64; ++c) { const float v = SRC[t * TLD + c]; s += v; s2 += v * v; }
    const float mu = s * (1.f / 64.f);
    const float rs = rsqrtf(s2 * (1.f / 64.f) - mu * mu + EPSf);
    for (int c = 0; c < 64; ++c)
      SRC[t * TLD + c] = (SRC[t * TLD + c] - mu) * rs * ln1g[c] + ln1b[c];
  }
  __syncthreads();
  if (threadIdx.x < 128) {   // FFN (64->8->64, exact GELU) + LN2 + writeout
    const int t = threadIdx.x;
    float h1[8];
#pragma unroll
    for (int f = 0; f < 8; ++f) {
      float a = l1b[f];
      for (int c = 0; c < 64; ++c) a += SRC[t * TLD + c] * l1w[c * 8 + f];
      h1[f] = 0.5f * a * (1.f + erff(a * 0.70710678118654752f));
    }
    for (int c = 0; c < 64; ++c) {
      float a = l2b[c];
#pragma unroll
      for (int f = 0; f < 8; ++f) a += h1[f] * l2w[f * 64 + c];
      OT[t * TLD + c] = SRC[t * TLD + c] + a;
    }
    float s = 0.f, s2 = 0.f;
    for (int c = 0; c < 64; ++c) { const float v = OT[t * TLD + c]; s += v; s2 += v * v; }
    const float mu = s * (1.f / 64.f);
    const float rs = rsqrtf(s2 * (1.f / 64.f) - mu * mu + EPSf);
    for (int c = 0; c < 64; ++c) {
      const float v = (OT[t * TLD + c] - mu) * rs * ln2g[c] + ln2b[c];
      yt[((((b << 6) + c) << 8) | k) * 128 + t] = v;
    }
  }
}

// ---------------- final channel FFN: one block per (b, 256-column slab) ----------------
#define FLD 65
#define HLD 129
__global__ __launch_bounds__(256) void k_ffn(const float* yin, const float* w1, const float* b1,
                                             const float* w2, const float* b2, float* outp) {
  __shared__ float XF[256 * FLD];
  __shared__ float HB[256 * HLD];
  const int b = blockIdx.x >> 7;
  const int s0 = (blockIdx.x & 127) << 8;
  for (int i = threadIdx.x; i < 64 * 256; i += 256) {
    const int c = i >> 8, n = i & 255;
    XF[n * FLD + c] = yin[(((b << 6) + c) << 15) + s0 + n];
  }
  __syncthreads();
  gemm_block(256, 128, 64,
    [&](int tm, int k0, int ln) { return frag_a(XF, FLD, tm, k0, ln); },
    [&](int tn, int k0, int ln) { return frag_b(w1, 128, tn, k0, ln); },
    [&](int tm, int tn, v8f av, int ln) {
      const int n = ln & 15, mh = (ln >> 4) << 3;
#pragma unroll
      for (int r = 0; r < 8; ++r)
        HB[(tm + mh + r) * HLD + tn + n] = fmaxf(av[r] + b1[tn + n], 0.f);
    });
  __syncthreads();
  gemm_block(256, 64, 128,
    [&](int tm, int k0, int ln) { return frag_a(HB, HLD, tm, k0, ln); },
    [&](int tn, int k0, int ln) { return frag_b(w2, 64, tn, k0, ln); },
    [&](int tm, int tn, v8f av, int ln) {
      const int n = ln & 15, mh = (ln >> 4) << 3;
#pragma unroll
      for (int r = 0; r < 8; ++r) {
        const int row = tm + mh + r, c = tn + n;
        outp[(((b << 6) + c) << 15) + s0 + row] = av[r] + b2[c] + XF[row * FLD + c];
      }
    });
}

// ---------------- elementwise + group norm ----------------
__global__ __launch_bounds__(256) void k_add2(const float* a, float* dst) {
  const int i = blockIdx.x * 256 + threadIdx.x;
  dst[i] += a[i];
}
__global__ __launch_bounds__(256) void k_add3(float* a, const float* b, const float* c) {
  const int i = blockIdx.x * 256 + threadIdx.x;
  a[i] = a[i] + b[i] + c[i];
}
__global__ __launch_bounds__(256) void k_gn_stats(const float* z, float* st) {
  const int blk = blockIdx.x;  // b*4 + group
  const float* base = z + (size_t)blk * 16 * 32768;
  float s = 0.f, s2 = 0.f;
  for (int i = threadIdx.x; i < 16 * 32768; i += 256) { const float v = base[i]; s += v; s2 += v * v; }
  __shared__ float r1[256], r2[256];
  r1[threadIdx.x] = s; r2[threadIdx.x] = s2; __syncthreads();
  for (int off = 128; off > 0; off >>= 1) {
    if (threadIdx.x < off) { r1[threadIdx.x] += r1[threadIdx.x + off]; r2[threadIdx.x] += r2[threadIdx.x + off]; }
    __syncthreads();
  }
  if (threadIdx.x == 0) {
    const float inv = 1.f / 524288.f;
    const float mu = r1[0] * inv;
    const float var = r2[0] * inv - mu * mu;
    st[blk * 2] = mu;
    st[blk * 2 + 1] = rsqrtf(var + EPSf);
  }
}
__global__ __launch_bounds__(256) void k_gn_apply(const float* z, const float* st, const float* gm,
                                                  const float* bt, float* outp) {
  const int i = blockIdx.x * 256 + threadIdx.x;
  const int c = (i >> 15) & 63;
  const int blk = (i >> 21) * 4 + (c >> 4);
  outp[i] = (z[i] - st[blk * 2]) * st[blk * 2 + 1] * gm[c] + bt[c];
}

// ---------------- host launcher ----------------
extern "C" void kernel_launch(void* const* d_in, const int* in_sizes, int n_in,
                              void* d_out, int out_size, void* d_ws, size_t ws_size,
                              hipStream_t stream) {
  (void)in_sizes; (void)n_in; (void)out_size; (void)ws_size;
  const float* y    = (const float*)d_in[0];
  const float* a1   = (const float*)d_in[1];
  const float* a2   = (const float*)d_in[2];
  const float* nv1  = (const float*)d_in[3];
  const float* nv2  = (const float*)d_in[4];
  const float* gw   = (const float*)d_in[5];
  const float* gb   = (const float*)d_in[6];
  const float* swq  = (const float*)d_in[7];
  const float* swk  = (const float*)d_in[8];
  const float* swv  = (const float*)d_in[9];
  const float* spk  = (const float*)d_in[10];
  const float* spv  = (const float*)d_in[11];
  const float* swo  = (const float*)d_in[12];
  const float* sbo  = (const float*)d_in[13];
  const float* twq  = (const float*)d_in[14];
  const float* twk  = (const float*)d_in[15];
  const float* twv  = (const float*)d_in[16];
  const float* tbq  = (const float*)d_in[17];
  const float* tbk  = (const float*)d_in[18];
  const float* tbv  = (const float*)d_in[19];
  const float* two  = (const float*)d_in[20];
  const float* tbo  = (const float*)d_in[21];
  const float* l1w  = (const float*)d_in[22];
  const float* l1b  = (const float*)d_in[23];
  const float* l2w  = (const float*)d_in[24];
  const float* l2b  = (const float*)d_in[25];
  const float* ln1g = (const float*)d_in[26];
  const float* ln1b = (const float*)d_in[27];
  const float* ln2g = (const float*)d_in[28];
  const float* ln2b = (const float*)d_in[29];
  const float* gnlg = (const float*)d_in[30];
  const float* gnlb = (const float*)d_in[31];
  const float* gnsg = (const float*)d_in[32];
  const float* gnsb = (const float*)d_in[33];
  const float* gntg = (const float*)d_in[34];
  const float* gntb = (const float*)d_in[35];
  const float* f1w  = (const float*)d_in[36];
  const float* f1b  = (const float*)d_in[37];
  const float* f2w  = (const float*)d_in[38];
  const float* f2b  = (const float*)d_in[39];
  const float* gn2g = (const float*)d_in[40];
  const float* gn2b = (const float*)d_in[41];

  float* W    = (float*)d_ws;
  float* N1   = W;                    // y_local path / y_in2
  float* N2   = W + NTOT;             // y_s path / ffn out
  float* N3   = W + 2 * NTOT;         // y_t path
  float* A1T  = W + 3 * NTOT;
  float* A2T  = A1T + 65536;
  float* ADPT = A2T + 65536;
  float* STAT = ADPT + 65536;

  const int EB = NTOT / 256;

  k_adp<<<256, 256, 0, stream>>>(nv1, nv2, ADPT);
  k_tr256<<<256, 256, 0, stream>>>(a1, A1T);
  k_tr256<<<256, 256, 0, stream>>>(a2, A2T);

  k_gcn<<<256, 256, 0, stream>>>(y, A1T, A2T, ADPT, gw, gb, N1);
  k_spatial<<<256, 256, 0, stream>>>(y, swq, swk, swv, spk, spv, swo, sbo, N2);
  k_temporal<<<512, 256, 0, stream>>>(y, twq, twk, twv, tbq, tbk, tbv, two, tbo,
                                      l1w, l1b, l2w, l2b, ln1g, ln1b, ln2g, ln2b, N3);

  // y_local = GN(y + raw)
  k_add2<<<EB, 256, 0, stream>>>(y, N1);
  k_gn_stats<<<8, 256, 0, stream>>>(N1, STAT);
  k_gn_apply<<<EB, 256, 0, stream>>>(N1, STAT, gnlg, gnlb, N1);
  // y_s
  k_add2<<<EB, 256, 0, stream>>>(y, N2);
  k_gn_stats<<<8, 256, 0, stream>>>(N2, STAT);
  k_gn_apply<<<EB, 256, 0, stream>>>(N2, STAT, gnsg, gnsb, N2);
  // y_t
  k_add2<<<EB, 256, 0, stream>>>(y, N3);
  k_gn_stats<<<8, 256, 0, stream>>>(N3, STAT);
  k_gn_apply<<<EB, 256, 0, stream>>>(N3, STAT, gntg, gntb, N3);

  // y_in2 = y_local + y_s + y_t  (in place into N1)
  k_add3<<<EB, 256, 0, stream>>>(N1, N2, N3);
  // h2 + y_in2 -> N2
  k_ffn<<<256, 256, 0, stream>>>(N1, f1w, f1b, f2w, f2b, N2);
  // final group norm -> d_out
  k_gn_stats<<<8, 256, 0, stream>>>(N2, STAT);
  k_gn_apply<<<EB, 256, 0, stream>>>(N2, STAT, gn2g, gn2b, (float*)d_out);
}